// Loss_all_15221364097647
// MI455X (gfx1250) — compile-verified
//
#include <hip/hip_runtime.h>
#include <hip/hip_bf16.h>

#define BATCH   128
#define NPARAM  62
#define NPTS    200     // 68 keypoints + 132 resampled
#define NROWS   600     // 3*NPTS
#define NROWS_P 608     // padded to 38*16
#define KDIM    50
#define KPAD    52      // padded to 13*4
#define MAGIC   0.00057339936f
#define EPSW    1e-6f

typedef __attribute__((ext_vector_type(2))) float v2f;
typedef __attribute__((ext_vector_type(8))) float v8f;

// ---- workspace layout (floats) ----
#define O_PARAM   0                             // 128*62
#define O_PARAMG  (O_PARAM  + BATCH*NPARAM)     // 128*62
#define O_ALP     (O_PARAMG + BATCH*NPARAM)     // 128*52 zero-padded alpha (input)
#define O_ALPG    (O_ALP    + BATCH*KPAD)       // 128*52 zero-padded alpha (target)
#define O_WK      (O_ALPG   + BATCH*KPAD)       // 608*52
#define O_WV      (O_WK     + NROWS_P*KPAD)     // 608*52
#define O_UK      (O_WV     + NROWS_P*KPAD)     // 608 (zero-padded)
#define O_UV      (O_UK     + NROWS_P)          // 608 (zero-padded)
#define O_CN      (O_UV     + NROWS_P)          // 64 (w_col_norm)
#define O_T       (O_CN     + 64)               // 3 * 608 * 128 (padded rows writable)
#define O_ACC     (O_T      + 3*NROWS_P*BATCH)  // 8 accumulators
#define WS_FLOATS (O_ACC + 8)

// ------- param = input*std + mean ; padded alpha operands ; zero accumulators -------
__global__ void k_param(const float* __restrict__ input, const float* __restrict__ target,
                        const float* __restrict__ pmean, const float* __restrict__ pstd,
                        float* __restrict__ ws) {
    int i = blockIdx.x * blockDim.x + threadIdx.x;
    if (i < BATCH * NPARAM) {
        int c = i % NPARAM;
        ws[O_PARAM  + i] = input[i]  * pstd[c] + pmean[c];
        ws[O_PARAMG + i] = target[i] * pstd[c] + pmean[c];
    }
    if (i < BATCH * KPAD) {                      // B-operand staging, zero-padded K
        int n = i / KPAD, k = i % KPAD;
        float a  = 0.0f, ag = 0.0f;
        if (k < KDIM) {
            int c = 12 + k;
            a  = input[n * NPARAM + c]  * pstd[c] + pmean[c];
            ag = target[n * NPARAM + c] * pstd[c] + pmean[c];
        }
        ws[O_ALP  + i] = a;
        ws[O_ALPG + i] = ag;
    }
    if (i < 8) ws[O_ACC + i] = 0.0f;
}

// ---------------- gather basis rows for kmix (sel=0) / kmix_v (sel=1) ----------------
__global__ void k_gather(const float* __restrict__ u, const float* __restrict__ w_shp,
                         const float* __restrict__ w_exp, const int* __restrict__ keyindex,
                         const int* __restrict__ ridx_w, const int* __restrict__ ridx_v,
                         float* __restrict__ ws) {
    int r   = blockIdx.x;            // 0..607
    int t   = threadIdx.x;           // 0..63
    int sel = blockIdx.y;            // 0: kmix, 1: kmix_v
    float* Wd = ws + (sel ? O_WV : O_WK);
    float* Ud = ws + (sel ? O_UV : O_UK);
    const int* ridx = sel ? ridx_v : ridx_w;
    if (r >= NROWS) {                // zero padding rows (keeps GEMM guard-free)
        if (t < KPAD) Wd[r*KPAD + t] = 0.0f;
        if (t == 0)   Ud[r] = 0.0f;
        return;
    }
    int p = r / 3, c = r % 3;
    int vidx = (p < 68) ? keyindex[p] : ridx[p - 68];
    long grow = 3L * vidx + c;       // interleaved xyz row index
    if (t < 40)            Wd[r*KPAD + t] = w_shp[grow*40 + t];
    else if (t < KDIM)     Wd[r*KPAD + t] = w_exp[grow*10 + (t - 40)];
    else if (t < KPAD)     Wd[r*KPAD + t] = 0.0f;
    if (t == 0) Ud[r] = u[grow];
}

// ---------------- column norms of [wshp_b | wexp_b] over kmix rows ----------------
__global__ void k_colnorm(float* __restrict__ ws) {
    int j = threadIdx.x;             // 64 threads, 50 active
    if (j < KDIM) {
        float s = 0.0f;
        const float* Wk = ws + O_WK;
        for (int r = 0; r < NROWS; ++r) { float x = Wk[r*KPAD + j]; s += x * x; }
        ws[O_CN + j] = sqrtf(s);
    } else if (j < 64) {
        ws[O_CN + j] = 0.0f;
    }
}

// ---------------- T[which] = W @ Alpha^T + u  via V_WMMA_F32_16X16X4_F32 ----------------
// which 0: Wk x alpha(input)   (weights branch, tmpv)
// which 1: Wv x alpha(input)   (vdc pred + nwl pred)
// which 2: Wv x alpha(target)  (vdc gt + nwl gt)
__global__ __launch_bounds__(32) void k_gemm_wmma(const float* __restrict__ ws_c,
                                                  float* __restrict__ ws) {
    int which = blockIdx.z;
    int mt = blockIdx.y;             // 0..37
    int nt = blockIdx.x;             // 0..7
    const float* W = ws_c + ((which == 0) ? O_WK : O_WV);
    const float* P = ws_c + ((which == 2) ? O_ALPG : O_ALP);
    const float* U = ws_c + ((which == 0) ? O_UK : O_UV);

    int lane = threadIdx.x;          // 0..31
    int half = lane >> 4;            // K-half (A and B)
    int lr   = lane & 15;            // M (A) / N (B) / N (C,D)
    const float* Arow = W + (mt * 16 + lr) * KPAD + 2 * half;  // padded: no guards
    const float* Brow = P + (nt * 16 + lr) * KPAD + 2 * half;  // padded: no guards

    v8f c = {};
    #pragma unroll
    for (int kt = 0; kt < 13; ++kt) {
        // A 16x4 f32: VGPR = K-LSB, lane-half = K-MSB. B 4x16 mirrored, N along lanes.
        // Both 8B-aligned (KPAD even, offsets even) -> single b64 loads.
        v2f a = *(const v2f*)(Arow + kt * 4);
        v2f b = *(const v2f*)(Brow + kt * 4);
        c = __builtin_amdgcn_wmma_f32_16x16x4_f32(false, a, false, b, (short)0, c,
                                                  false, false);
    }
    // D 16x16 f32: row = vgpr + 8*lane-half, col = lane&15 ; add u in epilogue.
    // T slabs are 608 rows and U is zero-padded, so padded-row stores are safe:
    // no guards -> straight-line add+store epilogue.
    float* T = ws + O_T + which * NROWS_P * BATCH + (nt * 16 + lr);
    #pragma unroll
    for (int v = 0; v < 8; ++v) {
        int row = mt * 16 + v + 8 * half;
        T[row * BATCH] = c[v] + U[row];
    }
}

// ---------------- weights + loss_wpdc partial ----------------
__global__ __launch_bounds__(256) void k_wpdc(const float* __restrict__ input,
                                              const float* __restrict__ target,
                                              float* __restrict__ ws) {
    int n = blockIdx.x, t = threadIdx.x;
    const float* Tw = ws + O_T;                 // which = 0 slab
    const float* pp = ws + O_PARAM  + n * NPARAM;
    const float* pg = ws + O_PARAMG + n * NPARAM;
    __shared__ float nrm[3];
    __shared__ float wsh[64];
    __shared__ float mx;
    __shared__ float red[256];

    if (t < 3) nrm[t] = 0.0f;
    __syncthreads();
    float s0 = 0.f, s1 = 0.f, s2 = 0.f;
    for (int r = t; r < NROWS; r += 256) {
        float x = Tw[r * BATCH + n];
        int c = r % 3;
        if (c == 0) s0 += x * x; else if (c == 1) s1 += x * x; else s2 += x * x;
    }
    atomicAdd(&nrm[0], s0); atomicAdd(&nrm[1], s1); atomicAdd(&nrm[2], s2);
    __syncthreads();
    if (t < 3) nrm[t] = sqrtf(nrm[t]);
    __syncthreads();

    float w = 0.0f;
    if (t < NPARAM) {
        float pd = fabsf(pp[t] - pg[t]);
        if (t < 11) {
            int col = t & 3;                    // arange(11)%4
            float scale = (col < 3) ? nrm[col] : sqrtf(200.0f);  // offset_norm
            w = pd * scale + EPSW;
        } else if (t == 11) {
            w = EPSW;                           // zeros column + EPS (before max)
        } else {
            w = MAGIC * pd * ws[O_CN + (t - 12)] + EPSW;
        }
    }
    if (t < 64) wsh[t] = (t < NPARAM) ? w : 0.0f;
    __syncthreads();
    if (t == 0) {
        float m = wsh[0];
        for (int i = 1; i < NPARAM; ++i) m = fmaxf(m, wsh[i]);
        mx = m;
    }
    __syncthreads();

    float contrib = 0.0f;
    if (t < NPARAM) {
        float wn = (t == 11) ? 0.0f : (w / mx);
        float d = input[n * NPARAM + t] - target[n * NPARAM + t];
        contrib = wn * d * d;
    }
    red[t] = contrib;
    __syncthreads();
    for (int s = 128; s > 0; s >>= 1) { if (t < s) red[t] += red[t + s]; __syncthreads(); }
    if (t == 0) atomicAdd(&ws[O_ACC + 0], red[0]);
}

// ---------------- loss_vdc: rotate, offset, 200x200 chamfer ----------------
__global__ __launch_bounds__(256) void k_vdc(float* __restrict__ ws) {
    int n = blockIdx.x, t = threadIdx.x;
    const float* Tv  = ws + O_T + 1 * NROWS_P * BATCH;   // pred base verts
    const float* Tvg = ws + O_T + 2 * NROWS_P * BATCH;   // gt base verts
    __shared__ float pp[12], pg[12];
    __shared__ float X[NPTS][3];     // gt points
    __shared__ float Y[NPTS][3];     // pred points
    __shared__ float red[256];
    if (t < 12) { pp[t] = ws[O_PARAM + n*NPARAM + t]; pg[t] = ws[O_PARAMG + n*NPARAM + t]; }
    __syncthreads();
    for (int p = t; p < NPTS; p += 256) {
        float b0 = Tv [(3*p+0)*BATCH + n], b1 = Tv [(3*p+1)*BATCH + n], b2 = Tv [(3*p+2)*BATCH + n];
        float g0 = Tvg[(3*p+0)*BATCH + n], g1 = Tvg[(3*p+1)*BATCH + n], g2 = Tvg[(3*p+2)*BATCH + n];
        #pragma unroll
        for (int i = 0; i < 3; ++i) {
            Y[p][i] = pp[i*4+0]*b0 + pp[i*4+1]*b1 + pp[i*4+2]*b2 + pp[i*4+3];
            X[p][i] = pg[i*4+0]*g0 + pg[i*4+1]*g1 + pg[i*4+2]*g2 + pg[i*4+3];
        }
    }
    __syncthreads();
    float msum = 0.0f;
    if (t < NPTS) {
        float x0 = X[t][0], x1 = X[t][1], x2 = X[t][2];
        float y0 = Y[t][0], y1 = Y[t][1], y2 = Y[t][2];
        float bestA = 3.4e38f, bestB = 3.4e38f;
        for (int q = 0; q < NPTS; ++q) {
            float d0 = x0 - Y[q][0], d1 = x1 - Y[q][1], d2 = x2 - Y[q][2];
            bestA = fminf(bestA, d0*d0 + d1*d1 + d2*d2);      // min over pred per gt pt
            float e0 = X[q][0] - y0, e1 = X[q][1] - y1, e2 = X[q][2] - y2;
            bestB = fminf(bestB, e0*e0 + e1*e1 + e2*e2);      // min over gt per pred pt
        }
        msum = bestA + bestB;
    }
    red[t] = msum;
    __syncthreads();
    for (int s = 128; s > 0; s >>= 1) { if (t < s) red[t] += red[t + s]; __syncthreads(); }
    if (t == 0) atomicAdd(&ws[O_ACC + 1], red[0]);
}

// ---------------- loss_nwl: 68 landmarks, factored normals reduction ----------------
__global__ __launch_bounds__(128) void k_nwl(const float* __restrict__ normals,
                                             float* __restrict__ ws) {
    int n = blockIdx.x, t = threadIdx.x;
    const float* Tv  = ws + O_T + 1 * NROWS_P * BATCH;   // rows 0..203 = keypoint rows
    const float* Tvg = ws + O_T + 2 * NROWS_P * BATCH;
    __shared__ float pp[12], pg[12];
    __shared__ float red[128];
    if (t < 12) { pp[t] = ws[O_PARAM + n*NPARAM + t]; pg[t] = ws[O_PARAMG + n*NPARAM + t]; }
    __syncthreads();
    float local = 0.0f;
    if (t < 68) {
        int j = t;
        float b0 = Tv [(3*j+0)*BATCH + n], b1 = Tv [(3*j+1)*BATCH + n], b2 = Tv [(3*j+2)*BATCH + n];
        float g0 = Tvg[(3*j+0)*BATCH + n], g1 = Tvg[(3*j+1)*BATCH + n], g2 = Tvg[(3*j+2)*BATCH + n];
        float D = 0.0f;
        #pragma unroll
        for (int i = 0; i < 3; ++i) {
            float v  = pp[i*4+0]*b0 + pp[i*4+1]*b1 + pp[i*4+2]*b2 + pp[i*4+3];
            float vg = pg[i*4+0]*g0 + pg[i*4+1]*g1 + pg[i*4+2]*g2 + pg[i*4+3];
            if (i == 1) { v = 121.0f - v; vg = 121.0f - vg; }   // STD_SIZE + 1 flip
            float d = vg - v;
            D += d * d;                                          // sum over k of (68x3 diff^2)
        }
        float cs = 0.0f;                                         // column sum of normals
        for (int i = 0; i < 68; ++i) cs += normals[(n*68 + i)*68 + j];
        local = cs * D;
    }
    red[t] = local;
    __syncthreads();
    for (int s = 64; s > 0; s >>= 1) { if (t < s) red[t] += red[t + s]; __syncthreads(); }
    if (t == 0) atomicAdd(&ws[O_ACC + 2], red[0]);
}

// ---------------- finalize ----------------
__global__ void k_finalize(const float* __restrict__ ws, float* __restrict__ out) {
    float wpdc = ws[O_ACC + 0] / (float)(BATCH * NPARAM);
    float vdc  = (ws[O_ACC + 1] / (float)(BATCH * NPTS)) * 0.001f;
    float nwl  = (ws[O_ACC + 2] / (float)(BATCH * 68 * 3)) * 0.001f;
    out[0] = wpdc + 3.0f * vdc + 3.0f * nwl;
}

extern "C" void kernel_launch(void* const* d_in, const int* in_sizes, int n_in,
                              void* d_out, int out_size, void* d_ws, size_t ws_size,
                              hipStream_t stream) {
    (void)in_sizes; (void)n_in; (void)out_size; (void)ws_size;
    const float* input    = (const float*)d_in[0];
    const float* target   = (const float*)d_in[1];
    const float* normals  = (const float*)d_in[2];
    const float* pmean    = (const float*)d_in[3];
    const float* pstd     = (const float*)d_in[4];
    const float* u        = (const float*)d_in[5];
    const float* w_shp    = (const float*)d_in[6];
    const float* w_exp    = (const float*)d_in[7];
    const int*   keyindex = (const int*)d_in[8];
    const int*   ridx_w   = (const int*)d_in[9];
    const int*   ridx_v   = (const int*)d_in[10];
    float* ws  = (float*)d_ws;
    float* out = (float*)d_out;

    // 1. params + padded alpha operands + zero accumulators
    k_param<<<(BATCH * NPARAM + 255) / 256, 256, 0, stream>>>(input, target, pmean, pstd, ws);
    // 2. gather padded basis rows for kmix / kmix_v
    k_gather<<<dim3(NROWS_P, 2), 64, 0, stream>>>(u, w_shp, w_exp, keyindex, ridx_w, ridx_v, ws);
    // 3. column norms of gathered [w_shp|w_exp]
    k_colnorm<<<1, 64, 0, stream>>>(ws);
    // 4. three (600x50)@(50x128) GEMMs on the f32 WMMA pipe (+u epilogue)
    k_gemm_wmma<<<dim3(BATCH / 16, NROWS_P / 16, 3), 32, 0, stream>>>(ws, ws);
    // 5-7. loss branches
    k_wpdc<<<BATCH, 256, 0, stream>>>(input, target, ws);
    k_vdc<<<BATCH, 256, 0, stream>>>(ws);
    k_nwl<<<BATCH, 128, 0, stream>>>(normals, ws);
    // 8. combine
    k_finalize<<<1, 1, 0, stream>>>(ws, out);
}